// PurePytorchMHCLayer_37512244363479
// MI455X (gfx1250) — compile-verified
//
#include <hip/hip_runtime.h>
#include <hip/hip_bf16.h>
#include <stdint.h>

#define HIDDEN 4096
#define NEXP 4
#define SINKHORN_ITERS 20
#define EPS_F 1e-5f
#define LDSW (HIDDEN + 16)   // padded stream row stride (floats): banks 0/16/32/48

typedef float v2f __attribute__((ext_vector_type(2)));
typedef float v8f __attribute__((ext_vector_type(8)));

__device__ __forceinline__ float sig_(float v) { return 1.0f / (1.0f + expf(-v)); }

// Low 32 bits of an LDS pointer == DS byte address (AS(3) pointers are 32-bit).
__device__ __forceinline__ unsigned lds_lo32(const void* p) {
  return (unsigned)(size_t)(const __attribute__((address_space(3))) char*)p;
}

// ---------------------------------------------------------------------------
// Prep: Sinkhorn(4x4) + sigmoid gates. Trivial scalar work, one thread.
// P layout: [0..15] M, [16..19] sigmoid(H_pre), [20..23] 2*sigmoid(H_post)
// ---------------------------------------------------------------------------
__global__ void mhc_prep(const float* __restrict__ Hpre,
                         const float* __restrict__ Hpost,
                         const float* __restrict__ Hres,
                         float* __restrict__ P) {
  if (threadIdx.x != 0 || blockIdx.x != 0) return;
  float M[NEXP][NEXP];
  for (int i = 0; i < NEXP; ++i)
    for (int j = 0; j < NEXP; ++j)
      M[i][j] = expf(Hres[i * NEXP + j]);
  for (int it = 0; it < SINKHORN_ITERS; ++it) {
    for (int i = 0; i < NEXP; ++i) {               // row normalize
      float rs = M[i][0] + M[i][1] + M[i][2] + M[i][3];
      float inv = 1.0f / fmaxf(rs, EPS_F);
      for (int j = 0; j < NEXP; ++j) M[i][j] *= inv;
    }
    for (int j = 0; j < NEXP; ++j) {               // col normalize
      float cs = M[0][j] + M[1][j] + M[2][j] + M[3][j];
      float inv = 1.0f / fmaxf(cs, EPS_F);
      for (int i = 0; i < NEXP; ++i) M[i][j] *= inv;
    }
  }
  for (int i = 0; i < NEXP; ++i)
    for (int j = 0; j < NEXP; ++j) P[i * NEXP + j] = M[i][j];
  for (int k = 0; k < NEXP; ++k) P[16 + k] = sig_(Hpre[k]);
  for (int k = 0; k < NEXP; ++k) P[20 + k] = 2.0f * sig_(Hpost[k]);
}

// ---------------------------------------------------------------------------
// Main streaming kernel: one block per row b. 256 threads = 8 wave32.
// Dynamic LDS: xs[4][LDSW] (staged x row) | ys[HIDDEN] (y_norm) | red[1]
// ---------------------------------------------------------------------------
__global__ __launch_bounds__(256) void mhc_main(const float* __restrict__ x,
                                                const __hip_bfloat16* __restrict__ w,
                                                const float* __restrict__ P,
                                                float* __restrict__ out) {
  extern __shared__ float smem[];
  float* xs  = smem;                 // [4][LDSW], 64B row padding
  float* ys  = smem + 4 * LDSW;      // [HIDDEN]
  float* red = smem + 4 * LDSW + HIDDEN;

  const int t = threadIdx.x;
  const int b = blockIdx.x;
  const float* xg = x + (size_t)b * (NEXP * HIDDEN);

  if (t == 0) red[0] = 0.0f;

  // ---- Stage x[b] (64KB) into LDS with CDNA5 async global->LDS copies ----
  {
    unsigned lbase = lds_lo32(xs);
    unsigned long long gbase = (unsigned long long)xg;
#pragma unroll
    for (int j = 0; j < NEXP; ++j) {
      unsigned lrow = lbase + (unsigned)(j * LDSW) * 4u;
      unsigned long long grow = gbase + (unsigned long long)(j * HIDDEN) * 4ull;
#pragma unroll
      for (int k = 0; k < 4; ++k) {
        unsigned byte_off = (unsigned)(k * 256 + t) * 16u;   // per-lane b128 chunk
        unsigned long long g = grow + byte_off;
        unsigned l = lrow + byte_off;
        asm volatile("global_load_async_to_lds_b128 %0, %1, off"
                     :: "v"(l), "v"(g) : "memory");
      }
    }
    asm volatile("s_wait_asynccnt 0x0" ::: "memory");
  }
  __syncthreads();

  // ---- Uniform parameters (SMEM path) ----
  float Mm[16];
#pragma unroll
  for (int i = 0; i < 16; ++i) Mm[i] = P[i];
  float hpre[4], hpost[4];
#pragma unroll
  for (int k = 0; k < 4; ++k) { hpre[k] = P[16 + k]; hpost[k] = P[20 + k]; }

  // ---- Pass 1: pre-mix aggregate, bf16 round-trip, sum of squares ----
  // Thread t owns columns {4t..4t+3} + 1024*q, q=0..3 (lane-consecutive float4s)
  float aggbf[16];
  float ss = 0.0f;
#pragma unroll
  for (int q = 0; q < 4; ++q) {
    int c0 = q * 1024 + t * 4;
    float4 x0 = *(const float4*)(xs + 0 * LDSW + c0);
    float4 x1 = *(const float4*)(xs + 1 * LDSW + c0);
    float4 x2 = *(const float4*)(xs + 2 * LDSW + c0);
    float4 x3 = *(const float4*)(xs + 3 * LDSW + c0);
    float4 a;
    a.x = hpre[0]*x0.x + hpre[1]*x1.x + hpre[2]*x2.x + hpre[3]*x3.x;
    a.y = hpre[0]*x0.y + hpre[1]*x1.y + hpre[2]*x2.y + hpre[3]*x3.y;
    a.z = hpre[0]*x0.z + hpre[1]*x1.z + hpre[2]*x2.z + hpre[3]*x3.z;
    a.w = hpre[0]*x0.w + hpre[1]*x1.w + hpre[2]*x2.w + hpre[3]*x3.w;
    a.x = __bfloat162float(__float2bfloat16(a.x));
    a.y = __bfloat162float(__float2bfloat16(a.y));
    a.z = __bfloat162float(__float2bfloat16(a.z));
    a.w = __bfloat162float(__float2bfloat16(a.w));
    ss += a.x*a.x + a.y*a.y + a.z*a.z + a.w*a.w;
    aggbf[q*4+0] = a.x; aggbf[q*4+1] = a.y; aggbf[q*4+2] = a.z; aggbf[q*4+3] = a.w;
  }
  // wave32 reduce + one ds_add_f32 per wave
#pragma unroll
  for (int off = 16; off > 0; off >>= 1) ss += __shfl_xor(ss, off, 32);
  if ((t & 31) == 0) atomicAdd(red, ss);
  __syncthreads();

  float inv_rms = rsqrtf(red[0] * (1.0f / HIDDEN) + EPS_F);

  // ---- Pass 2: y_norm = aggbf * inv_rms * w, into LDS for the mix pass ----
#pragma unroll
  for (int q = 0; q < 4; ++q) {
    int c0 = q * 1024 + t * 4;
    float4 y4;
    y4.x = aggbf[q*4+0] * inv_rms * __bfloat162float(w[c0 + 0]);
    y4.y = aggbf[q*4+1] * inv_rms * __bfloat162float(w[c0 + 1]);
    y4.z = aggbf[q*4+2] * inv_rms * __bfloat162float(w[c0 + 2]);
    y4.w = aggbf[q*4+3] * inv_rms * __bfloat162float(w[c0 + 3]);
    *(float4*)(ys + c0) = y4;
  }
  __syncthreads();

  float* outb = out + (size_t)b * (NEXP * HIDDEN);

#if __has_builtin(__builtin_amdgcn_wmma_f32_16x16x4_f32)
  // ---- Pass 3: stream mix via V_WMMA_F32_16X16X4_F32 (full f32 precision) --
  // D(16x16) = A(16x4) * B(4x16) + 0 (SRC2 = inline 0).  A rows 0..3 = M;
  // B = 16-col tile of the 4 streams (conflict-free via 64B row padding).
  // hpost[i]*y_norm is added as 4 FMAs in the 16-lane store block.
  // 16x4 f32 A layout: lanes 0-15 = rows, VGPR0 = {K0 | K2}, VGPR1 = {K1 | K3}.
  {
    const int lane = t & 31;
    const int wid  = t >> 5;
    const bool hi  = lane >= 16;
    const int  nl  = lane & 15;

    float a0 = 0.0f, a1 = 0.0f;
    if (nl == 0)      { a0 = hi ? Mm[2]  : Mm[0];  a1 = hi ? Mm[3]  : Mm[1];  }
    else if (nl == 1) { a0 = hi ? Mm[6]  : Mm[4];  a1 = hi ? Mm[7]  : Mm[5];  }
    else if (nl == 2) { a0 = hi ? Mm[10] : Mm[8];  a1 = hi ? Mm[11] : Mm[9];  }
    else if (nl == 3) { a0 = hi ? Mm[14] : Mm[12]; a1 = hi ? Mm[15] : Mm[13]; }
    v2f A; A[0] = a0; A[1] = a1;

    const int j0 = hi ? 2 : 0, j1 = hi ? 3 : 1;   // B rows for this half-wave
    for (int g = 0; g < 32; ++g) {                // 8 waves * 32 * 16 cols = 4096
      int c = (wid << 9) + (g << 4) + nl;
      v2f B; B[0] = xs[j0 * LDSW + c]; B[1] = xs[j1 * LDSW + c];
      float yt = ys[c];                            // unconditional (broadcast)
      v8f Cz = {0.f,0.f,0.f,0.f,0.f,0.f,0.f,0.f};  // folds to inline SRC2=0
      v8f D = __builtin_amdgcn_wmma_f32_16x16x4_f32(
          /*neg_a=*/false, A, /*neg_b=*/false, B,
          /*c_mod=*/(short)0, Cz, /*reuse_a=*/false, /*reuse_b=*/false);
      if (!hi) {   // D VGPR i, lanes 0-15 = output stream i, N = lane
        outb[0 * HIDDEN + c] = D[0] + hpost[0] * yt;
        outb[1 * HIDDEN + c] = D[1] + hpost[1] * yt;
        outb[2 * HIDDEN + c] = D[2] + hpost[2] * yt;
        outb[3 * HIDDEN + c] = D[3] + hpost[3] * yt;
      }
    }
  }
#else
  // ---- VALU fallback: 4x4 mix with float4 b128 stores ----
#pragma unroll
  for (int q = 0; q < 4; ++q) {
    int c0 = q * 1024 + t * 4;
    float4 x0 = *(const float4*)(xs + 0 * LDSW + c0);
    float4 x1 = *(const float4*)(xs + 1 * LDSW + c0);
    float4 x2 = *(const float4*)(xs + 2 * LDSW + c0);
    float4 x3 = *(const float4*)(xs + 3 * LDSW + c0);
    float4 y4 = *(const float4*)(ys + c0);
#pragma unroll
    for (int i = 0; i < NEXP; ++i) {
      float m0 = Mm[i*4+0], m1 = Mm[i*4+1], m2 = Mm[i*4+2], m3 = Mm[i*4+3];
      float hp = hpost[i];
      float4 o;
      o.x = m0*x0.x + m1*x1.x + m2*x2.x + m3*x3.x + hp*y4.x;
      o.y = m0*x0.y + m1*x1.y + m2*x2.y + m3*x3.y + hp*y4.y;
      o.z = m0*x0.z + m1*x1.z + m2*x2.z + m3*x3.z + hp*y4.z;
      o.w = m0*x0.w + m1*x1.w + m2*x2.w + m3*x3.w + hp*y4.w;
      *(float4*)(outb + (size_t)i * HIDDEN + c0) = o;
    }
  }
#endif
}

// ---------------------------------------------------------------------------
extern "C" void kernel_launch(void* const* d_in, const int* in_sizes, int n_in,
                              void* d_out, int out_size, void* d_ws, size_t ws_size,
                              hipStream_t stream) {
  const float*          x     = (const float*)d_in[0];
  const __hip_bfloat16* w     = (const __hip_bfloat16*)d_in[1];
  const float*          Hpre  = (const float*)d_in[2];
  const float*          Hpost = (const float*)d_in[3];
  const float*          Hres  = (const float*)d_in[4];
  float* P = (float*)d_ws;     // 24 floats of prepped parameters

  mhc_prep<<<1, 32, 0, stream>>>(Hpre, Hpost, Hres, P);

  size_t smem = (size_t)(4 * LDSW + HIDDEN + 16) * sizeof(float);  // ~82KB LDS
  mhc_main<<<4096, 256, smem, stream>>>(x, w, P, (float*)d_out);
  (void)in_sizes; (void)n_in; (void)out_size; (void)ws_size;
}